// MagNorm48_29557964931355
// MI455X (gfx1250) — compile-verified
//
#include <hip/hip_runtime.h>
#include <stdint.h>

// ---------------- problem constants (from reference setup_inputs) ----------
#define Bn 32
#define Tn 4096
#define Fn 481
#define ALPHA_F 0.99f
#define OMA_F   0.01f        // 1 - ALPHA
#define EPS_F   1e-12f
#define VAR0_F  1600.0f      // 40^2

// ---------------- tiling -----------------------------------------------
// One channel per lane (wave32), single-wave workgroups.
// TDM stages [T_TILE x 32ch] fp32 tiles (16 KB) into a 4-deep LDS ring.
// 3 tiles in flight per wave * 512 waves = ~24 MB outstanding -> above the
// 23.3 TB/s x ~700ns latency-BW product; per-lane loads (LOADcnt<=63, ~8 KB
// per wave) cannot get there, and the 1924-B row stride defeats B128 loads.
#define CH_PER_WAVE 32
#define T_TILE 128
#define NBUF 4
#define NTILES (Tn / T_TILE)   // 32

typedef uint32_t u32;
typedef uint64_t u64;
typedef u32 uint4v __attribute__((ext_vector_type(4)));
typedef int  int8v __attribute__((ext_vector_type(8)));
typedef int  int4v __attribute__((ext_vector_type(4)));

// Issue one Tensor-DMA load of a [T_TILE rows x 32 cols] fp32 tile with row
// stride Fn elements, from global 'gaddr' to LDS byte offset 'lds_byte'.
// D# layout per CDNA5 ISA ch.8 (group0: count/lds/global/type, group1: dims).
__device__ __forceinline__ void tdm_load_tile(u64 gaddr, u32 lds_byte) {
  uint4v g0;
  g0.x = 1u;                                 // count=1 (valid), gather off
  g0.y = lds_byte;                           // D#.lds_addr (bytes)
  g0.z = (u32)gaddr;                         // D#.global_addr[31:0]
  g0.w = (u32)(gaddr >> 32) | 0x80000000u;   // global_addr[56:32] | type=2
  int8v g1;
  g1[0] = 0x00020000;                        // data_size=2 -> 4-byte elems
  g1[1] = 0;                                 // abar=0 | tensor_dim0 lo16 (td0=0x40000000)
  g1[2] = 0x00004000;                        // td0 hi16 | tensor_dim1 lo16
  g1[3] = (CH_PER_WAVE << 16) | 0x4000;      // tile_dim0=32 | td1 hi16
  g1[4] = T_TILE;                            // tile_dim1=128, tile_dim2=0 (2D)
  g1[5] = Fn;                                // tensor_dim0_stride = 481 elems
  g1[6] = 0;
  g1[7] = 0;
  int4v gz4 = {0, 0, 0, 0};                  // groups 2/3: tile_dim3=0 -> unused
  int8v gz8 = {0, 0, 0, 0, 0, 0, 0, 0};      // trailing group (clang-23 form)
#if __has_builtin(__builtin_amdgcn_tensor_load_to_lds)
  // amdgpu-toolchain (clang-23) 6-arg form:
  // (uint32x4 g0, int32x8 g1, int32x4, int32x4, int32x8, i32 cpol)
  __builtin_amdgcn_tensor_load_to_lds(g0, g1, gz4, gz4, gz8, 0);
#else
  asm volatile("tensor_load_to_lds %0, %1, %2, %3"
               :: "s"(g0), "s"(g1), "s"(gz4), "s"(gz4) : "memory");
#endif
}

#if __has_builtin(__builtin_amdgcn_s_wait_tensorcnt)
#define WAIT_TENSOR(n) do { __builtin_amdgcn_s_wait_tensorcnt(n); \
                            asm volatile("" ::: "memory"); } while (0)
#else
#define WAIT_TENSOR(n) asm volatile("s_wait_tensorcnt %0" :: "i"(n) : "memory")
#endif

__global__ __launch_bounds__(CH_PER_WAVE)
void magnorm_ema_kernel(const float* __restrict__ x,
                        const float* __restrict__ mu0,
                        float* __restrict__ out)
{
  __shared__ float lds[NBUF * T_TILE * CH_PER_WAVE];   // 64 KB ring

  const int lane   = threadIdx.x;                 // 0..31, wave32
  const int chBase = blockIdx.x * CH_PER_WAVE;    // f-tile base
  const int b      = blockIdx.y;
  const int ch     = chBase + lane;
  const bool valid = (ch < Fn);

  // Per-wave tile origin in x: x[b, t0, chBase]. Rows strided by Fn floats.
  const u64 xbase = (u64)(uintptr_t)x +
                    4ull * ((u64)b * (u64)Tn * (u64)Fn + (u64)chBase);
  const u64 tileStride = 4ull * (u64)T_TILE * (u64)Fn;

  float mu  = valid ? mu0[(size_t)b * Fn + ch] : 0.0f;
  float var = VAR0_F;

  float* outp = out + ((size_t)b * Tn * Fn + ch);

  const u32 ldsBase  = (u32)(uintptr_t)(&lds[0]);  // LDS byte offset (addr[31:0])
  const u32 bufBytes = (u32)(T_TILE * CH_PER_WAVE * 4);

  // Prologue: keep NBUF-1 = 3 tiles in flight.
  tdm_load_tile(xbase,                    ldsBase);
  tdm_load_tile(xbase +      tileStride,  ldsBase +     bufBytes);
  tdm_load_tile(xbase + 2 *  tileStride,  ldsBase + 2 * bufBytes);

  int buf = 0, nbuf = NBUF - 1;
  for (int ib = 0; ib < NTILES; ++ib) {
    if (ib + (NBUF - 1) < NTILES) {
      // Reissue into the ring slot whose previous readers are already
      // consumed (in-order issue + register deps force prior ds_loads done).
      tdm_load_tile(xbase + (u64)(ib + NBUF - 1) * tileStride,
                    ldsBase + (u32)nbuf * bufBytes);
      WAIT_TENSOR(NBUF - 1);     // tensor ops complete in order per wave
    } else if (ib == NTILES - 3) {
      WAIT_TENSOR(2);
    } else if (ib == NTILES - 2) {
      WAIT_TENSOR(1);
    } else {
      WAIT_TENSOR(0);
    }

    const float* bufp = &lds[buf * (T_TILE * CH_PER_WAVE) + lane];
    float* orow = outp + (size_t)ib * T_TILE * Fn;
#pragma unroll 8
    for (int r = 0; r < T_TILE; ++r) {
      float xt = bufp[r * CH_PER_WAVE];            // ds_load, conflict-free
      mu  = fmaf(ALPHA_F, mu, OMA_F * xt);         // mu = a*mu + (1-a)*x
      float d = xt - mu;                           // uses updated mu (ref order)
      var = fmaf(ALPHA_F, var, (OMA_F * d) * d);   // var = a*var + (1-a)*d^2
      float s = __builtin_amdgcn_sqrtf(var) + EPS_F;
      float y = d * __builtin_amdgcn_rcpf(s);
      if (valid) __builtin_nontemporal_store(y, orow);  // streaming store
      orow += Fn;
    }
    buf  = (buf  == NBUF - 1) ? 0 : buf  + 1;
    nbuf = (nbuf == NBUF - 1) ? 0 : nbuf + 1;
  }
}

extern "C" void kernel_launch(void* const* d_in, const int* in_sizes, int n_in,
                              void* d_out, int out_size, void* d_ws, size_t ws_size,
                              hipStream_t stream) {
  const float* x   = (const float*)d_in[0];   // [B, T, F] fp32
  const float* mu0 = (const float*)d_in[1];   // [B, F] fp32
  float* out = (float*)d_out;                 // [B, T, F] fp32
  (void)in_sizes; (void)n_in; (void)out_size; (void)d_ws; (void)ws_size;

  dim3 grid((Fn + CH_PER_WAVE - 1) / CH_PER_WAVE, Bn);  // (16, 32)
  magnorm_ema_kernel<<<grid, CH_PER_WAVE, 0, stream>>>(x, mu0, out);
}